// ModelfreeGCN_45801531244835
// MI455X (gfx1250) — compile-verified
//
#include <hip/hip_runtime.h>
#include <stdint.h>

#define TPB   256   // threads per block (8 wave32)
#define TILE  256   // edges staged per block
#define GRP   16    // lanes per edge: 16 lanes x float4 = 64 floats
#define D_FEAT 64

// CDNA5 async global->LDS copy path (gfx1250). Guarded so both device and
// host compilation passes stay legal; falls back to plain loads otherwise.
#if defined(__AMDGCN__) &&                                            \
    __has_builtin(__builtin_amdgcn_global_load_async_to_lds_b64) &&   \
    __has_builtin(__builtin_amdgcn_global_load_async_to_lds_b32) &&   \
    __has_builtin(__builtin_amdgcn_s_wait_asynccnt)
#define USE_ASYNC_LDS 1
typedef int v2i __attribute__((ext_vector_type(2)));
typedef __attribute__((address_space(1))) v2i glb_v2i;
typedef __attribute__((address_space(3))) v2i lds_v2i;
typedef __attribute__((address_space(1))) int glb_i32;
typedef __attribute__((address_space(3))) int lds_i32;
#else
#define USE_ASYNC_LDS 0
#endif

// ---------------- Phase 1: degree = 1 (self loop) + scatter(w) at col -------
__global__ void gcn_deg_init(float* __restrict__ deg, int N) {
  int i = blockIdx.x * blockDim.x + threadIdx.x;
  if (i < N) deg[i] = 1.0f;  // self-loop fill weight
}

__global__ void gcn_deg_accum(const long long* __restrict__ col,
                              const float* __restrict__ w,
                              float* __restrict__ deg, int E) {
  int e = blockIdx.x * blockDim.x + threadIdx.x;
  if (e < E) atomicAdd(&deg[(int)col[e]], w[e]);
}

// ---------------- Phase 2: dinv = rsqrt(deg) in place -----------------------
__global__ void gcn_deg_to_dinv(float* __restrict__ deg, int N) {
  int i = blockIdx.x * blockDim.x + threadIdx.x;
  if (i < N) {
    float d = deg[i];
    deg[i] = d > 0.0f ? rsqrtf(d) : 0.0f;
  }
}

// ---------------- Phase 3: out[i] = x[i] * dinv[i]^2 (self-loop term) -------
__global__ void gcn_self_loop(const float4* __restrict__ x4,
                              const float* __restrict__ dinv,
                              float4* __restrict__ out4, int n4) {
  int idx = blockIdx.x * blockDim.x + threadIdx.x;
  if (idx < n4) {
    float s = dinv[idx >> 4];        // 16 float4 per node -> broadcast load
    float nrm = s * s;
    float4 v = x4[idx];
    v.x *= nrm; v.y *= nrm; v.z *= nrm; v.w *= nrm;
    out4[idx] = v;
  }
}

// ---------------- Phase 4: out[col] += x[row] * (dinv[row]*w*dinv[col]) -----
__global__ __launch_bounds__(TPB)
void gcn_edge_scatter(const float* __restrict__ x,
                      const float* __restrict__ ew,
                      const long long* __restrict__ row_idx,
                      const long long* __restrict__ col_idx,
                      const float* __restrict__ dinv,
                      float* __restrict__ out, int E) {
  __shared__ long long s_r64[TILE];
  __shared__ long long s_c64[TILE];
  __shared__ float     s_w[TILE];
  __shared__ int       s_r[TILE];
  __shared__ int       s_c[TILE];
  __shared__ float     s_n[TILE];

  const int t = threadIdx.x;
  const long long base = (long long)blockIdx.x * TILE;
  if (base >= E) return;
  const long long rem = E - base;
  const int n = rem < TILE ? (int)rem : TILE;

  // ---- stage edge tile into LDS (async on CDNA5) ----
  if (t < n) {
    long long e = base + t;
#if USE_ASYNC_LDS
    __builtin_amdgcn_global_load_async_to_lds_b64(
        (glb_v2i*)(uintptr_t)(row_idx + e), (lds_v2i*)&s_r64[t], 0, 0);
    __builtin_amdgcn_global_load_async_to_lds_b64(
        (glb_v2i*)(uintptr_t)(col_idx + e), (lds_v2i*)&s_c64[t], 0, 0);
    __builtin_amdgcn_global_load_async_to_lds_b32(
        (glb_i32*)(uintptr_t)(ew + e), (lds_i32*)&s_w[t], 0, 0);
#else
    s_r64[t] = row_idx[e];
    s_c64[t] = col_idx[e];
    s_w[t]   = ew[e];
#endif
  }
#if USE_ASYNC_LDS
  __builtin_amdgcn_s_wait_asynccnt(0);
#endif
  __syncthreads();

  // ---- per-edge norm ----
  if (t < n) {
    int r = (int)s_r64[t];
    int c = (int)s_c64[t];
    s_r[t] = r;
    s_c[t] = c;
    s_n[t] = s_w[t] * dinv[r] * dinv[c];
  }
  __syncthreads();

  // ---- scatter: 16 lanes per edge, each lane owns one float4 of the row ----
  const int g = t >> 4;   // edge group within tile (0..15)
  const int l = t & 15;   // feature quarter-lane (0..15)
  for (int e = g; e < n; e += TPB / GRP) {
    const int e2 = e + TPB / GRP;
    if (e2 < n)  // hide random-gather latency of the next edge's source row
      __builtin_prefetch(x + (long long)s_r[e2] * D_FEAT + l * 4, 0, 0);

    const float nrm = s_n[e];
    const float4 xv =
        *(const float4*)(x + (long long)s_r[e] * D_FEAT + l * 4);
    float* o = out + (long long)s_c[e] * D_FEAT + l * 4;
    atomicAdd(o + 0, xv.x * nrm);
    atomicAdd(o + 1, xv.y * nrm);
    atomicAdd(o + 2, xv.z * nrm);
    atomicAdd(o + 3, xv.w * nrm);
  }
}

extern "C" void kernel_launch(void* const* d_in, const int* in_sizes, int n_in,
                              void* d_out, int out_size, void* d_ws, size_t ws_size,
                              hipStream_t stream) {
  const float*     x   = (const float*)d_in[0];
  const float*     ew  = (const float*)d_in[1];
  const long long* ei  = (const long long*)d_in[2];  // int64 [2, E]
  const int E = in_sizes[1];
  const int N = in_sizes[0] / D_FEAT;
  const long long* row = ei;       // sources
  const long long* col = ei + E;   // targets
  float* out = (float*)d_out;
  float* deg = (float*)d_ws;       // N floats; becomes dinv in place

  const int bN = (N + TPB - 1) / TPB;
  const int bE = (E + TPB - 1) / TPB;

  gcn_deg_init   <<<bN, TPB, 0, stream>>>(deg, N);
  gcn_deg_accum  <<<bE, TPB, 0, stream>>>(col, ew, deg, E);
  gcn_deg_to_dinv<<<bN, TPB, 0, stream>>>(deg, N);

  const int n4 = N * (D_FEAT / 4);
  gcn_self_loop<<<(n4 + TPB - 1) / TPB, TPB, 0, stream>>>(
      (const float4*)x, deg, (float4*)out, n4);

  const int tiles = (E + TILE - 1) / TILE;
  gcn_edge_scatter<<<tiles, TPB, 0, stream>>>(x, ew, row, col, deg, out, E);
}